// TriPlaneEncoder_52725018525994
// MI455X (gfx1250) — compile-verified
//
#include <hip/hip_runtime.h>
#include <cmath>

#define ENC_LEVELS 16
#define ENC_MAX_RES 2048
#define LUT_CAP 8192   // entries; 16KB as u16. >= 7393 used entries.

#if defined(__HIP_DEVICE_COMPILE__)
#if __has_builtin(__builtin_amdgcn_global_load_async_to_lds_b128)
#define HAVE_ASYNC_LD 1
#endif
#if __has_builtin(__builtin_amdgcn_global_store_async_from_lds_b128)
#define HAVE_ASYNC_ST 1
#endif
#endif

struct LevelTab {
  float resm1[ENC_LEVELS];     // (float)(res - 1)
  float resf [ENC_LEVELS];     // (float)res
  int   offs [ENC_LEVELS + 1]; // cumulative (res+1); offs[16] = total entries
};

typedef float floatx4 __attribute__((ext_vector_type(4)));
typedef int   v4i     __attribute__((vector_size(16)));
typedef __attribute__((address_space(1))) v4i* gv4i_ptr;  // global
typedef __attribute__((address_space(3))) v4i* lv4i_ptr;  // LDS

// ---------------------------------------------------------------------------
// Setup: build the (level, ci) -> c_ori index LUT once into workspace.
// lutv = (u16)(int)((float)ci / res * 2047)  -- identical float semantics to
// the reference's  (c / res * (MAX_RES-1)).astype(int32).
// ---------------------------------------------------------------------------
__global__ __launch_bounds__(256) void lut_build_kernel(unsigned short* __restrict__ ws_lut,
                                                        LevelTab lt)
{
  const int total = lt.offs[ENC_LEVELS];
  for (int e = threadIdx.x; e < LUT_CAP; e += 256) {
    unsigned short v = 0;
    if (e < total) {
      int l = 0;
      #pragma unroll
      for (int k = 1; k < ENC_LEVELS; ++k) l = (e >= lt.offs[k]) ? k : l;
      const int ci = e - lt.offs[l];
      v = (unsigned short)(int)((float)ci / lt.resf[l] * 2047.0f);
    }
    ws_lut[e] = v;
  }
}

// ---------------------------------------------------------------------------
// Main encoder.
// ---------------------------------------------------------------------------
__global__ __launch_bounds__(256) void triplane_enc_kernel(
    const float* __restrict__ positions,
    const float* __restrict__ plane_table,
    float* __restrict__ out,
    const unsigned short* __restrict__ ws_lut,
    LevelTab lt, int B, int use_ws)
{
  __shared__ unsigned short lut[LUT_CAP];
  __shared__ floatx4 stage[256 * 9];   // padded stride 9 -> ~2-way bank conflicts

  const int tid = threadIdx.x;

  if (use_ws) {
#if defined(HAVE_ASYNC_LD)
    // CDNA5 async DMA: pull the 16KB LUT (hot in L2) straight into LDS,
    // no VGPR round-trip. 4 x b128 per thread covers 256*4*16B = 16KB.
    #pragma unroll
    for (int k = 0; k < 4; ++k) {
      const int ofs = (tid + k * 256) * 16;
      __builtin_amdgcn_global_load_async_to_lds_b128(
          (gv4i_ptr)((char*)ws_lut + ofs),
          (lv4i_ptr)((char*)lut + ofs),
          0, 0);
    }
    // ASYNCcnt is per-wave: drain before the barrier so other waves see data.
#if __has_builtin(__builtin_amdgcn_s_wait_asynccnt)
    __builtin_amdgcn_s_wait_asynccnt(0);
#else
    asm volatile("s_wait_asynccnt 0x0" ::: "memory");
#endif
#else
    for (int e = tid; e < LUT_CAP; e += 256) lut[e] = ws_lut[e];
#endif
  } else {
    // Fallback: compute LUT in-block (only if workspace is too small).
    const int total = lt.offs[ENC_LEVELS];
    for (int e = tid; e < total; e += 256) {
      int l = 0;
      #pragma unroll
      for (int k = 1; k < ENC_LEVELS; ++k) l = (e >= lt.offs[k]) ? k : l;
      const int ci = e - lt.offs[l];
      lut[e] = (unsigned short)(int)((float)ci / lt.resf[l] * 2047.0f);
    }
  }
  __syncthreads();

  const int b = blockIdx.x * 256 + tid;
  const bool active = (b < B);

  if (active) {
    // Streamed once -> non-temporal; keep the 96MB table resident in L2.
    const float c0 = __builtin_nontemporal_load(positions + 3 * (size_t)b + 0);
    const float c1 = __builtin_nontemporal_load(positions + 3 * (size_t)b + 1);
    const float c2 = __builtin_nontemporal_load(positions + 3 * (size_t)b + 2);
    const float crd[3] = {c0, c1, c2};   // plane p = (crd[p], crd[(p+1)%3])

    const float2* __restrict__ tab = reinterpret_cast<const float2*>(plane_table);

    float o0[ENC_LEVELS];
    float o1[ENC_LEVELS];

    float2 fcur[12], fnxt[12];
    float  frc[3], frn[3];   // per-coordinate fractional parts

    // Per-coordinate work done ONCE (each coordinate feeds two planes), then
    // issue all 12 gathers for level l.
    auto fetch = [&](int l, float2* f, float* fr, int* i0, int* i1) {
      const float r1  = lt.resm1[l];
      const int   off = lt.offs[l];
      #pragma unroll
      for (int c = 0; c < 3; ++c) {
        const float s = crd[c] * r1 + 0.5f;
        const float g = floorf(s);
        fr[c] = s - g;
        const int ci = (int)g;
        i0[c] = lut[off + ci];
        i1[c] = lut[off + ci + 1];
      }
      #pragma unroll
      for (int p = 0; p < 3; ++p) {
        const int q = (p + 1) % 3;            // dim-1 coordinate of plane p
        const int row0 = i0[q] * ENC_MAX_RES;
        const int row1 = i1[q] * ENC_MAX_RES;
        const float2* __restrict__ pl = tab + (size_t)p * (ENC_MAX_RES * ENC_MAX_RES);
        f[4 * p + 0] = pl[row0 + i0[p]];
        f[4 * p + 1] = pl[row0 + i1[p]];
        f[4 * p + 2] = pl[row1 + i0[p]];
        f[4 * p + 3] = pl[row1 + i1[p]];
      }
    };

    {
      int i0[3], i1[3];
      fetch(0, fcur, frc, i0, i1);
    }

    #pragma unroll
    for (int l = 0; l < ENC_LEVELS; ++l) {
      // Software pipeline: issue next level's 12 gathers before consuming
      // this level's -> ~24 loads in flight, partial loadcnt waits.
      if (l + 1 < ENC_LEVELS) {
        int i0[3], i1[3];
        fetch(l + 1, fnxt, frn, i0, i1);
      }

      float v0 = 1.0f, v1 = 1.0f;
      #pragma unroll
      for (int p = 0; p < 3; ++p) {
        const int q = (p + 1) % 3;
        const float wx1 = frc[p], wx0 = 1.0f - frc[p];
        const float wy1 = frc[q], wy0 = 1.0f - frc[q];
        const float w00 = wx0 * wy0;
        const float w10 = wx1 * wy0;
        const float w01 = wx0 * wy1;
        const float w11 = wx1 * wy1;
        const float2 f00 = fcur[4 * p + 0];
        const float2 f10 = fcur[4 * p + 1];
        const float2 f01 = fcur[4 * p + 2];
        const float2 f11 = fcur[4 * p + 3];
        // Reference accumulation order: corner 0,1,2,3
        float e0 = w00 * f00.x; e0 += w10 * f10.x; e0 += w01 * f01.x; e0 += w11 * f11.x;
        float e1 = w00 * f00.y; e1 += w10 * f10.y; e1 += w01 * f01.y; e1 += w11 * f11.y;
        v0 *= e0;
        v1 *= e1;
      }
      o0[l] = v0;
      o1[l] = v1;

      #pragma unroll
      for (int q = 0; q < 12; ++q) fcur[q] = fnxt[q];
      #pragma unroll
      for (int c = 0; c < 3; ++c) frc[c] = frn[c];
    }

    // Stage this thread's 32 floats (8 x float4 chunks) for the transpose.
    #pragma unroll
    for (int k = 0; k < 4; ++k) {
      floatx4 s0 = { o0[4 * k + 0], o0[4 * k + 1], o0[4 * k + 2], o0[4 * k + 3] };
      floatx4 s1 = { o1[4 * k + 0], o1[4 * k + 1], o1[4 * k + 2], o1[4 * k + 3] };
      stage[tid * 9 + k]     = s0;
      stage[tid * 9 + 4 + k] = s1;
    }
  }
  __syncthreads();

  // Fully-coalesced block store: wave-level stores now cover contiguous
  // full 128B lines instead of 32 partial lines each. NT so the 128MB output
  // stream doesn't evict the plane table from L2.
  floatx4* out4 = (floatx4*)out;
  const int base4 = blockIdx.x * 2048;      // 256 threads * 8 float4
  const int nf4   = B * 8;
  #pragma unroll
  for (int j = 0; j < 8; ++j) {
    const int m  = tid + j * 256;           // m = b_local*8 + chunk
    const int g4 = base4 + m;
    if (g4 < nf4) {
      const int sidx = (m >> 3) * 9 + (m & 7);
#if defined(HAVE_ASYNC_ST)
      // LDS -> global DMA store, cpol=1 -> TH_NT. ASYNCcnt drained by the
      // implicit wait-idle at S_ENDPGM.
      __builtin_amdgcn_global_store_async_from_lds_b128(
          (gv4i_ptr)(out4 + g4),
          (lv4i_ptr)&stage[sidx],
          0, 1);
#else
      __builtin_nontemporal_store(stage[sidx], out4 + g4);
#endif
    }
  }
}

extern "C" void kernel_launch(void* const* d_in, const int* in_sizes, int n_in,
                              void* d_out, int out_size, void* d_ws, size_t ws_size,
                              hipStream_t stream) {
  const float* positions   = (const float*)d_in[0];
  const float* plane_table = (const float*)d_in[1];
  float* out = (float*)d_out;

  const int B = in_sizes[0] / 3;

  // Replicate the Python double-precision level-resolution math on the host.
  LevelTab lt;
  const double log_b = std::log((double)ENC_MAX_RES / 16.0) / (double)(ENC_LEVELS - 1);
  int acc = 0;
  for (int l = 0; l < ENC_LEVELS; ++l) {
    const double scale = 16.0 * std::exp((double)l * log_b) - 1.0;
    const int res = (int)std::ceil(scale) + 1;
    lt.resm1[l] = (float)(res - 1);
    lt.resf[l]  = (float)res;
    lt.offs[l]  = acc;
    acc += res + 1;
  }
  lt.offs[ENC_LEVELS] = acc;  // == 7393 <= LUT_CAP

  const int use_ws = (ws_size >= (size_t)(LUT_CAP * sizeof(unsigned short)));
  unsigned short* ws_lut = (unsigned short*)d_ws;

  if (use_ws) {
    hipLaunchKernelGGL(lut_build_kernel, dim3(1), dim3(256), 0, stream, ws_lut, lt);
  }

  const int threads = 256;
  const int blocks = (B + threads - 1) / threads;
  hipLaunchKernelGGL(triplane_enc_kernel, dim3(blocks), dim3(threads), 0, stream,
                     positions, plane_table, out, ws_lut, lt, B, use_ws);
}